// TorchDiAcyclicGraph_18021682774643
// MI455X (gfx1250) — compile-verified
//
#include <hip/hip_runtime.h>

// Sum-product network over a binary tree: B=256 batch, D=1024 leaves, K=128 units.
// f32 end-to-end; matmuls on V_WMMA_F32_16X16X4_F32 (bandwidth-bound workload, so
// full-precision WMMA is free relative to the HBM roofline). Weight tiles are
// streamed into LDS by the Tensor Data Mover (tensor_load_to_lds, TENSORcnt),
// double-buffered against the WMMA loop.

#define HALF_LOG_2PI 0.9189385332046727f

static constexpr int BB = 256;   // batch
static constexpr int DD = 1024;  // leaves
static constexpr int KK = 128;   // units per node

typedef __attribute__((ext_vector_type(2))) float v2f;
typedef __attribute__((ext_vector_type(8))) float v8f;
typedef __attribute__((ext_vector_type(4))) unsigned int u32x4;
typedef __attribute__((ext_vector_type(8))) int i32x8;
typedef __attribute__((ext_vector_type(4))) int i32x4;

// P-chunk layout in LDS: 128 rows (i) x 32 floats (j) padded to 36-float stride.
static constexpr int PSTRIDE = 36;                 // floats per LDS row
static constexpr int PCHUNK_FLOATS = 128 * PSTRIDE;

// ---------------------------------------------------------------- leaf layer
__global__ __launch_bounds__(256)
void leaf_kernel(const float* __restrict__ x, const float* __restrict__ mu,
                 const float* __restrict__ ls, float* __restrict__ y) {
    int idx = blockIdx.x * 256 + threadIdx.x;           // over B*D*K
    int k = idx & (KK - 1);
    int d = (idx / KK) & (DD - 1);
    int b = idx / (KK * DD);
    float xs = x[b * DD + d];
    float m  = mu[d * KK + k];
    float l  = ls[d * KK + k];
    float z  = (xs - m) * __expf(-l);
    y[idx] = -0.5f * z * z - l - HALF_LOG_2PI;
}

// ------------------------------------------------- softmax of mixture weights
// One wave32 per 128-wide row: float4 per lane, shfl_xor reductions.
__global__ __launch_bounds__(256)
void softmax_kernel(const float* __restrict__ w, float* __restrict__ p, int nrows) {
    int wave = (blockIdx.x * 256 + threadIdx.x) >> 5;
    int lane = threadIdx.x & 31;
    if (wave >= nrows) return;
    const float4* row = reinterpret_cast<const float4*>(w + (size_t)wave * KK);
    float4 v = row[lane];
    float mx = fmaxf(fmaxf(v.x, v.y), fmaxf(v.z, v.w));
    #pragma unroll
    for (int off = 16; off > 0; off >>= 1) mx = fmaxf(mx, __shfl_xor(mx, off, 32));
    float e0 = __expf(v.x - mx), e1 = __expf(v.y - mx);
    float e2 = __expf(v.z - mx), e3 = __expf(v.w - mx);
    float s = e0 + e1 + e2 + e3;
    #pragma unroll
    for (int off = 16; off > 0; off >>= 1) s += __shfl_xor(s, off, 32);
    float inv = 1.0f / s;
    float4 o; o.x = e0 * inv; o.y = e1 * inv; o.z = e2 * inv; o.w = e3 * inv;
    reinterpret_cast<float4*>(p + (size_t)wave * KK)[lane] = o;
}

// -------------------------------------------- TDM descriptor for one P chunk
// Loads a [128 x 32]-f32 tile of the [128 x 128] node matrix (tile start =
// chunk_ptr, row stride 128 elems) into LDS at lds_byte_addr, padding each
// 32-DWORD row with 4 DWORDs -> effective LDS row stride of 36 floats.
__device__ __forceinline__ void tdm_load_p_chunk(const float* chunk_ptr,
                                                 unsigned int lds_byte_addr) {
    unsigned long long ga = (unsigned long long)(uintptr_t)chunk_ptr;
    u32x4 g0;
    g0.x = 1u;                                        // count=1, user descriptor
    g0.y = lds_byte_addr;                             // lds_addr
    g0.z = (unsigned int)(ga & 0xFFFFFFFFu);          // global_addr[31:0]
    g0.w = (unsigned int)((ga >> 32) & 0x1FFFFFFu)    // global_addr[56:32]
         | (2u << 30);                                // type = 2 ("image")
    i32x8 g1;
    g1[0] = (int)((2u << 16)        // data_size = 4 bytes
                | (1u << 20)        // pad_enable
                | (4u << 22)        // pad_interval: 32 DWORDs
                | (3u << 25));      // pad_amount:   4 DWORDs
    g1[1] = (int)(128u << 16);      // tensor_dim0 = 128 (elems), low 16 bits
    g1[2] = (int)(128u << 16);      // tensor_dim0 hi=0 | tensor_dim1 = 128 lo16
    g1[3] = (int)(32u  << 16);      // tensor_dim1 hi=0 | tile_dim0 = 32
    g1[4] = (int)(128u);            // tile_dim1 = 128 | tile_dim2 = 0
    g1[5] = (int)(128u);            // tensor_dim0_stride = 128 (elems), low 32
    g1[6] = 0;                      // stride hi | tensor_dim1_stride lo (unused)
    g1[7] = 0;
    i32x4 g2 = {0, 0, 0, 0};        // 2-D tensor: groups 2/3 unused
    i32x4 g3 = {0, 0, 0, 0};
    i32x8 g4 = {0, 0, 0, 0, 0, 0, 0, 0};   // 6-arg toolchain variant: unused group
    __builtin_amdgcn_tensor_load_to_lds(g0, g1, g2, g3, g4, 0);
}

// ---------------------------------------------------------------- tree level
// yin: [B, 2d, K]; P: [d, K, K] row-major (i,j), rows sum to 1; yout: [B, d, K].
// out[b,n,i] = m[b,n] + log( sum_j P[n,i,j] * exp(t[b,n,j]-m) ),
//   t[b,n,j] = yin[b,2n,j] + yin[b,2n+1,j],  m = max_j t.
// Block: 256 threads (8 waves), one node n x one 16-row batch tile. Wave w owns
// the 16x16 output tile at i in [16w,16w+16): 32x V_WMMA_F32_16X16X4_F32 with
// A = exp-tile rows (LDS, stride 132) and B[k][i] = P[i][j0+k] read straight
// from the TDM-staged, 36-float-stride LDS tile (bank-conflict free).
__global__ __launch_bounds__(256)
void level_kernel(const float* __restrict__ yin, const float* __restrict__ P,
                  float* __restrict__ yout, int d) {
    __shared__ float sE[16 * 132];                       // exp(t - m)
    __shared__ __align__(16) float sP[2][PCHUNK_FLOATS]; // TDM double buffer
    __shared__ float sM[16];
    __shared__ float sRed[256];

    const int n     = blockIdx.x % d;
    const int btile = blockIdx.x / d;
    const int tid   = threadIdx.x;
    const int wid   = tid >> 5;
    const float* Pn = P + (size_t)n * KK * KK;

    // Kick off chunk 0 DMA immediately; it overlaps with phase A below.
    if (wid == 0) {
        tdm_load_p_chunk(Pn, (unsigned int)(uintptr_t)(void*)&sP[0][0]);
    }

    // ---- phase A: fused child-sum, row max, exp into LDS -------------------
    const int row = tid >> 4;          // 0..15 batch row in tile
    const int kq  = (tid & 15) << 3;   // 8 consecutive k per thread
    const int b   = btile * 16 + row;
    const float* y0 = yin + ((size_t)b * (2 * d) + 2 * n) * KK;

    float4 a0 = reinterpret_cast<const float4*>(y0 + kq)[0];
    float4 a1 = reinterpret_cast<const float4*>(y0 + kq)[1];
    float4 c0 = reinterpret_cast<const float4*>(y0 + KK + kq)[0];
    float4 c1 = reinterpret_cast<const float4*>(y0 + KK + kq)[1];
    float t[8] = {a0.x + c0.x, a0.y + c0.y, a0.z + c0.z, a0.w + c0.w,
                  a1.x + c1.x, a1.y + c1.y, a1.z + c1.z, a1.w + c1.w};
    float lm = t[0];
    #pragma unroll
    for (int q = 1; q < 8; ++q) lm = fmaxf(lm, t[q]);
    sRed[tid] = lm;
    __syncthreads();
    if (tid < 16) {
        float m = sRed[tid * 16];
        #pragma unroll
        for (int q = 1; q < 16; ++q) m = fmaxf(m, sRed[tid * 16 + q]);
        sM[tid] = m;
    }
    __syncthreads();
    const float mrow = sM[row];
    #pragma unroll
    for (int q = 0; q < 8; ++q) sE[row * 132 + kq + q] = __expf(t[q] - mrow);

    // ---- phase B: WMMA over j, TDM double-buffered in 4 chunks of 32 -------
    const int lane  = tid & 31;
    const int i0    = wid << 4;                 // wave's i-tile base
    const int Mrow  = lane & 15;                // A-fragment row (batch)
    const int khalf = (lane >> 4) << 1;         // 0 (lanes 0-15) / 2 (16-31)
    const int icol  = i0 + (lane & 15);         // B-fragment column (i)

    v8f acc = {};
    #pragma unroll
    for (int jc = 0; jc < 4; ++jc) {
        if (wid == 0) {
            if (jc < 3) {
                tdm_load_p_chunk(Pn + (jc + 1) * 32,
                                 (unsigned int)(uintptr_t)(void*)&sP[(jc + 1) & 1][0]);
                __builtin_amdgcn_s_wait_tensorcnt(1);   // chunk jc landed
            } else {
                __builtin_amdgcn_s_wait_tensorcnt(0);
            }
        }
        __syncthreads();                                // chunk jc visible to all
        const float* sPc = &sP[jc & 1][0];
        #pragma unroll
        for (int kc = 0; kc < 8; ++kc) {
            const int j0 = kc * 4;
            v2f afrag, bfrag;
            afrag.x = sE[Mrow * 132 + jc * 32 + j0 + khalf];
            afrag.y = sE[Mrow * 132 + jc * 32 + j0 + khalf + 1];
            bfrag.x = sPc[icol * PSTRIDE + j0 + khalf];
            bfrag.y = sPc[icol * PSTRIDE + j0 + khalf + 1];
            acc = __builtin_amdgcn_wmma_f32_16x16x4_f32(
                false, afrag, false, bfrag, (short)0, acc, false, false);
        }
        __syncthreads();                                // buffer may be reused
    }

    // ---- epilogue: m + log(s), coalesced 16x16 tile stores -----------------
    const int nhalf = lane >> 4;
    const int ncol  = lane & 15;
    #pragma unroll
    for (int r = 0; r < 8; ++r) {
        const int Mr = r + nhalf * 8;
        const float val = sM[Mr] + __logf(acc[r]);
        const int bo = btile * 16 + Mr;
        yout[((size_t)bo * d + n) * KK + i0 + ncol] = val;
    }
}

// ------------------------------------------------------------------- driver
extern "C" void kernel_launch(void* const* d_in, const int* in_sizes, int n_in,
                              void* d_out, int out_size, void* d_ws, size_t ws_size,
                              hipStream_t stream) {
    const float* x  = (const float*)d_in[0];
    const float* mu = (const float*)d_in[1];
    const float* ls = (const float*)d_in[2];
    const float* w  = (const float*)d_in[3];
    float* out = (float*)d_out;

    char* ws = (char*)d_ws;
    const size_t yA_bytes = (size_t)BB * DD * KK * sizeof(float);   // 134 MB
    const size_t yB_bytes = yA_bytes / 2;                           //  67 MB
    float* bufA = (float*)ws;
    float* bufB = (float*)(ws + yA_bytes);
    float* Pbuf = (float*)(ws + yA_bytes + yB_bytes);               // (D-1)*K*K

    // 1) leaf log-densities -> bufA
    leaf_kernel<<<(BB * DD * KK) / 256, 256, 0, stream>>>(x, mu, ls, bufA);

    // 2) softmax all (D-1)*K weight rows -> Pbuf
    const int nrows = (DD - 1) * KK;
    softmax_kernel<<<(nrows + 7) / 8, 256, 0, stream>>>(w, Pbuf, nrows);

    // 3) 10 tree levels (product + log-space mixture), ping-pong buffers
    const float* yin = bufA;
    int d = DD, off = 0, flip = 0;
    while (d > 1) {
        d >>= 1;
        float* yout = (d == 1) ? out : (flip ? bufA : bufB);
        level_kernel<<<(BB / 16) * d, 256, 0, stream>>>(
            yin, Pbuf + (size_t)off * KK * KK, yout, d);
        yin = yout;
        off += d;
        flip ^= 1;
    }
}